// Denoising_Unet_32787780337928
// MI455X (gfx1250) — compile-verified
//
#include <hip/hip_runtime.h>
#include <hip/hip_bf16.h>

// ---------------- problem constants ----------------
#define NN     2048      // nodes
#define NE     65536     // edges per graph
#define HIDD   512       // hidden
#define NH     8         // heads
#define DHD    64        // head dim
#define OUTDIM 256
#define SLOPE  0.2f
#define LEPS   1e-5f

typedef float v2f __attribute__((ext_vector_type(2)));
typedef float v8f __attribute__((ext_vector_type(8)));
typedef unsigned int v4u __attribute__((ext_vector_type(4)));
typedef int v4i __attribute__((ext_vector_type(4)));
typedef int v8i __attribute__((ext_vector_type(8)));

__device__ __forceinline__ float lrelu(float x) { return x >= 0.f ? x : SLOPE * x; }

// monotonic float<->uint encoding for atomicMax on floats
__device__ __forceinline__ unsigned enc_f(float f) {
    unsigned u = __float_as_uint(f);
    return (u & 0x80000000u) ? ~u : (u | 0x80000000u);
}
__device__ __forceinline__ float dec_f(unsigned u) {
    return (u & 0x80000000u) ? __uint_as_float(u ^ 0x80000000u) : __uint_as_float(~u);
}

// ---------------- TDM: 2D tile DMA global -> LDS (ISA 8.3-8.5 D# packing) ----------------
// tile_x = contiguous elements per line (dwords), tile_y = lines,
// stride = element stride between lines, optional LDS write padding.
__device__ __forceinline__ void tdm_load_2d(unsigned lds_off, const void* gptr,
                                            unsigned tile_x, unsigned tile_y,
                                            unsigned long long stride_elems,
                                            unsigned pad_en, unsigned pad_int_f, unsigned pad_amt_f)
{
    unsigned long long ga = (unsigned long long)(uintptr_t)gptr;
    v4u g0;
    g0.x = 1u;                                   // count=1, is_restore=0, gather off
    g0.y = lds_off;                              // lds_addr (bytes)
    g0.z = (unsigned)ga;                         // global_addr[31:0]
    g0.w = (unsigned)(ga >> 32) | (2u << 30);    // global_addr[56:32] | type=2 (image)
    const unsigned td0 = tile_x, td1 = tile_y;   // tensor dims: exact fit, tile always in-bounds
    unsigned w0 = (2u << 16)                     // data_size = 4 bytes
                | (pad_en << 20) | (pad_int_f << 22) | (pad_amt_f << 25);
    unsigned w1 = (td0 & 0xFFFFu) << 16;                          // tensor_dim0[15:0]
    unsigned w2 = (td0 >> 16) | ((td1 & 0xFFFFu) << 16);          // dim0[31:16] | dim1[15:0]
    unsigned w3 = (td1 >> 16) | (tile_x << 16);                   // dim1[31:16] | tile_dim0
    unsigned w4 = tile_y & 0xFFFFu;                               // tile_dim1 (tile_dim2 = 0)
    unsigned w5 = (unsigned)(stride_elems & 0xFFFFFFFFull);       // tensor_dim0_stride[31:0]
    unsigned w6 = (unsigned)((stride_elems >> 32) & 0xFFFFull);   // stride[47:32]
    v8i g1 = { (int)w0, (int)w1, (int)w2, (int)w3, (int)w4, (int)w5, (int)w6, 0 };
    v4i z4 = { 0, 0, 0, 0 };
#if __clang_major__ >= 23
    v8i z8 = { 0, 0, 0, 0, 0, 0, 0, 0 };
    __builtin_amdgcn_tensor_load_to_lds(g0, g1, z4, z4, z8, 0);
#else
    __builtin_amdgcn_tensor_load_to_lds(g0, g1, z4, z4, 0);
#endif
}

// ---------------- tiny utility kernels ----------------
__global__ void fill_f32_kernel(float* p, float v, int n) {
    int t = blockIdx.x * 256 + threadIdx.x; if (t < n) p[t] = v;
}
__global__ void fill_u32_kernel(unsigned* p, unsigned v, int n) {
    int t = blockIdx.x * 256 + threadIdx.x; if (t < n) p[t] = v;
}
__global__ void add_kernel(float* x, const float* y, int n) {
    int t = blockIdx.x * 256 + threadIdx.x; if (t < n) x[t] += y[t];
}
__global__ void bias_add_kernel(float* x, const float* b, int n, int C) {
    int t = blockIdx.x * 256 + threadIdx.x; if (t < n) x[t] += b[t % C];
}
__global__ void avg_kernel(const float* a, const float* b, float* o, int n) {
    int t = blockIdx.x * 256 + threadIdx.x; if (t < n) o[t] = 0.5f * (a[t] + b[t]);
}

// ---------------- GCN pieces ----------------
__global__ void degree_kernel(const int* src, const int* dst, float* dout, float* din_, int e) {
    int t = blockIdx.x * 256 + threadIdx.x; if (t >= e) return;
    atomicAdd(&dout[src[t]], 1.f);
    atomicAdd(&din_[dst[t]], 1.f);
}
__global__ void rsqrt_clamp_kernel(float* p, int n) {
    int t = blockIdx.x * 256 + threadIdx.x; if (t < n) p[t] = rsqrtf(fmaxf(p[t], 1.f));
}
__global__ __launch_bounds__(256)
void gcn_scatter_kernel(const int* src, const int* dst, const float* xw,
                        const float* rs_out, float* agg) {
    int e = blockIdx.x;
    int s = src[e], d = dst[e];
    float rs = rs_out[s];
    const float* xr = xw + (size_t)s * NN;
    float* ar = agg + (size_t)d * NN;
    for (int j = threadIdx.x; j < NN; j += 256) atomicAdd(&ar[j], xr[j] * rs);
}
__global__ void gcn_finalize_kernel(float* agg, const float* rs_in, const float* b) {
    size_t t = (size_t)blockIdx.x * 256 + threadIdx.x;   // over NN*NN
    int i = (int)(t / NN), j = (int)(t % NN);
    agg[t] = agg[t] * rs_in[i] + b[j];
}
__global__ void symm_clip_kernel(const float* P, float* S0) {
    size_t t = (size_t)blockIdx.x * 256 + threadIdx.x;   // over NN*NN
    int i = (int)(t / NN), j = (int)(t % NN);
    float v = 0.5f * (P[(size_t)i * NN + j] + P[(size_t)j * NN + i]);
    S0[t] = fminf(fmaxf(v, -1e10f), 1e10f);
}

// ---------------- WMMA f32 GEMM with TDM double-buffered LDS staging ----------------
// 256 threads = 8 waves (4 row x 2 col); block tile 128x64, K-step 16.
// Wave tile 32x32 = 4 WMMA accumulators (2 A-frags x 2 B-frags per k-step).
// Wave 0 issues tensor_load_to_lds for tile k+1 while all waves compute tile k.
template<bool TB>
__global__ __launch_bounds__(256)
void gemm_tdm_kernel(const float* __restrict__ A, const float* __restrict__ B,
                     float* __restrict__ C, int M, int N, int K,
                     const float* __restrict__ bias, const float* __restrict__ resid,
                     const float* __restrict__ prelu_a)
{
    // A: 128 rows x 16 k, LDS pitch 18 (TDM pad: 2 dwords per 16-dword line)
    // B (TB=1): 64 rows x 16 k, pitch 18 (same padding)
    // B (TB=0): 16 k-lines x 64 n, pitch 80 (TDM pad: 16 dwords per 64-dword line)
    __shared__ float As[2][128 * 18];
    __shared__ float Bs[2][16 * 80];
    const int t    = threadIdx.x;
    const int lane = t & 31;
    const int wave = t >> 5;
    const int wm   = wave & 3;     // row sub-tile (32 rows each)
    const int wn   = wave >> 2;    // col half (32 cols each)
    const int lr   = lane & 15;
    const int lh   = lane >> 4;    // half-wave -> K offset pair
    const int rowBase = blockIdx.y * 128;
    const int colBase = blockIdx.x * 64;

    auto issue = [&](int p, int kb) {
        if (wave == 0) {
            tdm_load_2d((unsigned)(uintptr_t)&As[p][0], A + (size_t)rowBase * K + kb,
                        16u, 128u, (unsigned long long)K, 1u, 3u, 1u);       // 16dw lines + 2dw pad
            if (TB)
                tdm_load_2d((unsigned)(uintptr_t)&Bs[p][0], B + (size_t)colBase * K + kb,
                            16u, 64u, (unsigned long long)K, 1u, 3u, 1u);
            else
                tdm_load_2d((unsigned)(uintptr_t)&Bs[p][0], B + (size_t)kb * N + colBase,
                            64u, 16u, (unsigned long long)N, 1u, 5u, 15u);   // 64dw lines + 16dw pad
        }
    };

    v8f acc[2][2] = {};
    issue(0, 0);
    int p = 0;
    for (int kb = 0; kb < K; kb += 16, p ^= 1) {
        if (wave == 0) __builtin_amdgcn_s_wait_tensorcnt((short)0);
        __syncthreads();                       // publish tile p; all reads of tile p^1 done
        if (kb + 16 < K) issue(p ^ 1, kb + 16);
        const float* Ap = &As[p][0];
        const float* Bp = &Bs[p][0];
        #pragma unroll
        for (int kk = 0; kk < 16; kk += 4) {
            const int ko = kk + 2 * lh;
            v2f a0, a1, b0, b1;
            a0.x = Ap[(wm * 32 + lr) * 18 + ko];      a0.y = Ap[(wm * 32 + lr) * 18 + ko + 1];
            a1.x = Ap[(wm * 32 + 16 + lr) * 18 + ko]; a1.y = Ap[(wm * 32 + 16 + lr) * 18 + ko + 1];
            if (TB) {
                b0.x = Bp[(wn * 32 + lr) * 18 + ko];      b0.y = Bp[(wn * 32 + lr) * 18 + ko + 1];
                b1.x = Bp[(wn * 32 + 16 + lr) * 18 + ko]; b1.y = Bp[(wn * 32 + 16 + lr) * 18 + ko + 1];
            } else {
                b0.x = Bp[ko * 80 + wn * 32 + lr];        b0.y = Bp[(ko + 1) * 80 + wn * 32 + lr];
                b1.x = Bp[ko * 80 + wn * 32 + 16 + lr];   b1.y = Bp[(ko + 1) * 80 + wn * 32 + 16 + lr];
            }
            acc[0][0] = __builtin_amdgcn_wmma_f32_16x16x4_f32(false, a0, false, b0, (short)0, acc[0][0], false, false);
            acc[0][1] = __builtin_amdgcn_wmma_f32_16x16x4_f32(false, a0, false, b1, (short)0, acc[0][1], false, false);
            acc[1][0] = __builtin_amdgcn_wmma_f32_16x16x4_f32(false, a1, false, b0, (short)0, acc[1][0], false, false);
            acc[1][1] = __builtin_amdgcn_wmma_f32_16x16x4_f32(false, a1, false, b1, (short)0, acc[1][1], false, false);
        }
    }

    const float aslope = prelu_a ? *prelu_a : 0.f;
    const int c0 = colBase + wn * 32 + lr;
    const int c1 = c0 + 16;
    const float bb0 = bias ? bias[c0] : 0.f;
    const float bb1 = bias ? bias[c1] : 0.f;
    #pragma unroll
    for (int r = 0; r < 2; ++r) {
        #pragma unroll
        for (int i = 0; i < 8; ++i) {
            int row = rowBase + wm * 32 + r * 16 + lh * 8 + i;   // D layout: v[i] rows i / i+8
            size_t o0 = (size_t)row * N + c0;
            size_t o1 = (size_t)row * N + c1;
            float v0 = acc[r][0][i] + bb0;
            float v1 = acc[r][1][i] + bb1;
            if (resid) { v0 += resid[o0]; v1 += resid[o1]; }
            if (prelu_a) { v0 = v0 >= 0.f ? v0 : aslope * v0; v1 = v1 >= 0.f ? v1 : aslope * v1; }
            C[o0] = v0; C[o1] = v1;
        }
    }
}

// ---------------- LayerNorm + PReLU (row-wise, in place) ----------------
__global__ __launch_bounds__(256)
void ln_prelu_kernel(float* x, const float* g, const float* b, const float* a_ptr, int C) {
    __shared__ float red[16];
    float* xr = x + (size_t)blockIdx.x * C;
    float s = 0.f, s2 = 0.f;
    for (int j = threadIdx.x; j < C; j += 256) { float v = xr[j]; s += v; s2 += v * v; }
    for (int o = 16; o; o >>= 1) { s += __shfl_xor(s, o, 32); s2 += __shfl_xor(s2, o, 32); }
    int w = threadIdx.x >> 5, lane = threadIdx.x & 31;
    if (!lane) { red[w] = s; red[w + 8] = s2; }
    __syncthreads();
    if (threadIdx.x == 0) {
        float ts = 0.f, ts2 = 0.f;
        for (int i = 0; i < 8; ++i) { ts += red[i]; ts2 += red[i + 8]; }
        red[0] = ts; red[8] = ts2;
    }
    __syncthreads();
    float mu   = red[0] / C;
    float var  = red[8] / C - mu * mu;
    float rstd = rsqrtf(var + LEPS);
    float a    = *a_ptr;
    for (int j = threadIdx.x; j < C; j += 256) {
        float v = (xr[j] - mu) * rstd * g[j] + b[j];
        xr[j] = v >= 0.f ? v : a * v;
    }
}

// ---------------- GAT: attention logits ----------------
__global__ __launch_bounds__(256)
void el_er_kernel(const float* __restrict__ z, const float* __restrict__ al,
                  const float* __restrict__ ar, float* el, float* er) {
    int lane = threadIdx.x & 31, wave = threadIdx.x >> 5;
    int wg = blockIdx.x * 8 + wave;
    int n = wg >> 3, h = wg & 7;
    const float* zr = z + (size_t)n * HIDD + h * DHD;
    float2 zv = *(const float2*)&zr[lane * 2];
    float2 a1 = *(const float2*)&al[h * DHD + lane * 2];
    float2 a2 = *(const float2*)&ar[h * DHD + lane * 2];
    float pl = zv.x * a1.x + zv.y * a1.y;
    float pr = zv.x * a2.x + zv.y * a2.y;
    for (int o = 16; o; o >>= 1) { pl += __shfl_xor(pl, o, 32); pr += __shfl_xor(pr, o, 32); }
    if (!lane) { el[n * NH + h] = pl; er[n * NH + h] = pr; }
}
__global__ void maxel_kernel(const float* el, float* mx) {
    __shared__ float sm[256];
    int h = blockIdx.x;
    float m = -INFINITY;
    for (int n = threadIdx.x; n < NN; n += 256) m = fmaxf(m, el[n * NH + h]);
    sm[threadIdx.x] = m; __syncthreads();
    for (int s = 128; s; s >>= 1) {
        if (threadIdx.x < s) sm[threadIdx.x] = fmaxf(sm[threadIdx.x], sm[threadIdx.x + s]);
        __syncthreads();
    }
    if (!threadIdx.x) mx[h] = sm[0];
}

// ---------------- dense GAT, fused flash-style with WMMA ----------------
__global__ __launch_bounds__(256)
void gat_dense_attn_kernel(const float* __restrict__ z, const float* __restrict__ el,
                           const float* __restrict__ er, const float* __restrict__ mx,
                           const float* __restrict__ bias, float* __restrict__ out)
{
    const int lane = threadIdx.x & 31;
    const int wg   = blockIdx.x * 8 + (threadIdx.x >> 5);
    const int head = wg >> 7;
    const int vt   = wg & 127;
    const int lr   = lane & 15;
    const int lh   = lane >> 4;
    const int v    = vt * 16 + lr;
    const float er_v = er[v * NH + head];
    const float m    = lrelu(mx[head] + er_v);    // leaky monotone -> exact column max
    v8f acc[4] = {};
    float dpart = 0.f;
    const int colb = head * DHD + lr;
    for (int u0 = 0; u0 < NN; u0 += 4) {
        const int ua = u0 + 2 * lh;
        int upf = (ua + 64 < NN) ? ua + 64 : NN - 1;           // stream-ahead prefetch
        __builtin_prefetch(&z[(size_t)upf * HIDD + colb], 0, 1);
        float w0 = __expf(lrelu(el[ua * NH + head] + er_v) - m);
        float w1 = __expf(lrelu(el[(ua + 1) * NH + head] + er_v) - m);
        dpart += w0 + w1;
        v2f af; af.x = w0; af.y = w1;
        #pragma unroll
        for (int nt = 0; nt < 4; ++nt) {
            v2f bf;
            bf.x = z[(size_t)ua * HIDD + colb + nt * 16];
            bf.y = z[(size_t)(ua + 1) * HIDD + colb + nt * 16];
            acc[nt] = __builtin_amdgcn_wmma_f32_16x16x4_f32(false, af, false, bf, (short)0, acc[nt], false, false);
        }
    }
    float dtot = dpart + __shfl_xor(dpart, 16, 32);
    #pragma unroll
    for (int i = 0; i < 8; ++i) {
        int rr = i + 8 * lh;
        float dr = __shfl(dtot, rr, 32);
        int row = vt * 16 + rr;
        #pragma unroll
        for (int nt = 0; nt < 4; ++nt) {
            int col = head * DHD + nt * 16 + lr;
            out[(size_t)row * HIDD + col] = acc[nt][i] / dr + bias[col];
        }
    }
}

// ---------------- sparse GAT kernels ----------------
__global__ void edge_max_kernel(const int* src, const int* dst, const float* el,
                                const float* er, unsigned* emax, int n) {
    int t = blockIdx.x * 256 + threadIdx.x; if (t >= n) return;
    int e = t >> 3, h = t & 7;
    float x = lrelu(el[src[e] * NH + h] + er[dst[e] * NH + h]);
    atomicMax(&emax[dst[e] * NH + h], enc_f(x));
}
__global__ void emax_fix_kernel(unsigned* p, int n) {
    int t = blockIdx.x * 256 + threadIdx.x; if (t >= n) return;
    float f = dec_f(p[t]);
    if (isinf(f) && f < 0.f) f = 0.f;     // where(isfinite(emax), emax, 0)
    ((float*)p)[t] = f;
}
__global__ void edge_ee_kernel(const int* src, const int* dst, const float* el,
                               const float* er, const float* emax, float* ee, float* den, int n) {
    int t = blockIdx.x * 256 + threadIdx.x; if (t >= n) return;
    int e = t >> 3, h = t & 7;
    float x = lrelu(el[src[e] * NH + h] + er[dst[e] * NH + h]);
    float w = __expf(x - emax[dst[e] * NH + h]);
    ee[t] = w;
    atomicAdd(&den[dst[e] * NH + h], w);
}
__global__ void edge_alpha_kernel(const int* dst, float* ee, const float* den, int n) {
    int t = blockIdx.x * 256 + threadIdx.x; if (t >= n) return;
    int e = t >> 3, h = t & 7;
    ee[t] /= fmaxf(den[dst[e] * NH + h], 1e-9f);
}
__global__ __launch_bounds__(256)
void edge_scatter_kernel(const int* src, const int* dst, const float* alpha,
                         const float* z, float* out) {
    int e = blockIdx.x;
    const float* zr = z + (size_t)src[e] * HIDD;
    float* outr = out + (size_t)dst[e] * HIDD;
    #pragma unroll
    for (int r = 0; r < 2; ++r) {
        int idx = threadIdx.x + r * 256;
        float a = alpha[e * NH + (idx >> 6)];
        atomicAdd(&outr[idx], a * zr[idx]);
    }
}

// ---------------- output concat: [N,H,DH]+[N,H,DH] -> [N,H,2DH] ----------------
__global__ void concat_kernel(const float* hf_, const float* hd_, float* dst) {
    size_t t = (size_t)blockIdx.x * 256 + threadIdx.x;   // over NN*HIDD
    int n = (int)(t / HIDD), c = (int)(t % HIDD);
    int h = c / DHD, d = c % DHD;
    dst[(size_t)n * (2 * HIDD) + h * (2 * DHD) + d]       = hf_[t];
    dst[(size_t)n * (2 * HIDD) + h * (2 * DHD) + DHD + d] = hd_[t];
}

// ================= host-side orchestration =================
static inline void gemm(hipStream_t s, const float* A, const float* B, float* C,
                        int M, int N, int K, int transB,
                        const float* bias, const float* resid, const float* pa) {
    dim3 g(N / 64, M / 128), b(256);
    if (transB) gemm_tdm_kernel<true ><<<g, b, 0, s>>>(A, B, C, M, N, K, bias, resid, pa);
    else        gemm_tdm_kernel<false><<<g, b, 0, s>>>(A, B, C, M, N, K, bias, resid, pa);
}

struct MlpP { const float *inW,*inb,*r1W,*r1b,*lng,*lnb,*r2W,*r2b,*outW,*outb,*a1,*a2; };

static inline MlpP mlp_from(void* const* d_in, int base) {
    MlpP p;
    p.inW  = (const float*)d_in[base + 0];  p.inb  = (const float*)d_in[base + 1];
    p.r1W  = (const float*)d_in[base + 2];  p.r1b  = (const float*)d_in[base + 3];
    p.lng  = (const float*)d_in[base + 4];  p.lnb  = (const float*)d_in[base + 5];
    p.r2W  = (const float*)d_in[base + 6];  p.r2b  = (const float*)d_in[base + 7];
    p.outW = (const float*)d_in[base + 8];  p.outb = (const float*)d_in[base + 9];
    p.a1   = (const float*)d_in[base + 10]; p.a2   = (const float*)d_in[base + 11];
    return p;
}

static void run_mlp(hipStream_t s, const MlpP& p, const float* x, int din, int dhid, int dout,
                    float* t1, float* t2, float* t3, float* out) {
    gemm(s, x,  p.inW,  t1,  NN, dhid, din,  0, p.inb,  nullptr, nullptr);
    gemm(s, t1, p.r1W,  t2,  NN, dhid, dhid, 0, p.r1b,  nullptr, nullptr);
    ln_prelu_kernel<<<NN, 256, 0, s>>>(t2, p.lng, p.lnb, p.a1, dhid);
    gemm(s, t2, p.r2W,  t3,  NN, dhid, dhid, 0, p.r2b,  t1,      nullptr);
    gemm(s, t3, p.outW, out, NN, dout, dhid, 0, p.outb, nullptr, p.a2);
}

static void run_gat_dense(hipStream_t s, const float* x, void* const* d_in, int base,
                          float* z, float* el, float* er, float* mx, float* out) {
    const float* W  = (const float*)d_in[base + 0];
    const float* al = (const float*)d_in[base + 1];
    const float* ar = (const float*)d_in[base + 2];
    const float* b  = (const float*)d_in[base + 3];
    gemm(s, x, W, z, NN, HIDD, HIDD, 0, nullptr, nullptr, nullptr);
    el_er_kernel<<<NN * NH / 8, 256, 0, s>>>(z, al, ar, el, er);
    maxel_kernel<<<NH, 256, 0, s>>>(el, mx);
    gat_dense_attn_kernel<<<NH * 128 / 8, 256, 0, s>>>(z, el, er, mx, b, out);
}

static void run_gat_sparse(hipStream_t s, const int* src, const int* dst,
                           const float* z, const float* el, const float* er, const float* bgat,
                           float* emax, float* den, float* ee, float* out) {
    const int nb_nh = (NN * NH + 255) / 256;
    const int nb_eh = (NE * NH + 255) / 256;
    fill_u32_kernel<<<nb_nh, 256, 0, s>>>((unsigned*)emax, 0x007FFFFFu, NN * NH); // enc(-inf)
    edge_max_kernel<<<nb_eh, 256, 0, s>>>(src, dst, el, er, (unsigned*)emax, NE * NH);
    emax_fix_kernel<<<nb_nh, 256, 0, s>>>((unsigned*)emax, NN * NH);
    fill_f32_kernel<<<nb_nh, 256, 0, s>>>(den, 0.f, NN * NH);
    edge_ee_kernel<<<nb_eh, 256, 0, s>>>(src, dst, el, er, emax, ee, den, NE * NH);
    edge_alpha_kernel<<<nb_eh, 256, 0, s>>>(dst, ee, den, NE * NH);
    fill_f32_kernel<<<NN * HIDD / 256, 256, 0, s>>>(out, 0.f, NN * HIDD);
    edge_scatter_kernel<<<NE, 256, 0, s>>>(src, dst, ee, z, out);
    bias_add_kernel<<<NN * HIDD / 256, 256, 0, s>>>(out, bgat, NN * HIDD, HIDD);
}

// input indices (setup_inputs dict order, depth-first)
enum {
    I_GF = 0, I_GD, I_FMRI, I_DTI, I_TEMB,
    I_GCNF_W, I_GCNF_B, I_GCND_W, I_GCND_B,
    I_MT = 9, I_MM = 21, I_MOF = 33, I_MOD = 45,
    I_D0 = 57, I_D1 = 61, I_U0 = 65, I_U1 = 69
};

// workspace layout (floats)
static constexpr size_t OFF_RS    = 0;                       // 4*2048 rsqrt-degree
static constexpr size_t OFF_EL    = OFF_RS    + 8192;        // N*8
static constexpr size_t OFF_ER    = OFF_EL    + 16384;
static constexpr size_t OFF_MX    = OFF_ER    + 16384;       // 64
static constexpr size_t OFF_EMAX  = OFF_MX    + 64;          // N*8
static constexpr size_t OFF_DEN   = OFF_EMAX  + 16384;       // N*8
static constexpr size_t OFF_EE    = OFF_DEN   + 16384;       // E*8
static constexpr size_t OFF_XW    = OFF_EE    + 524288;      // N*N
static constexpr size_t OFF_FMRIC = OFF_XW    + 4194304;
static constexpr size_t OFF_DTIC  = OFF_FMRIC + 4194304;
static constexpr size_t OFF_P     = OFF_DTIC  + 4194304;
static constexpr size_t OFF_S0    = OFF_P     + 4194304;
static constexpr size_t OFF_T1    = OFF_S0    + 4194304;     // 2048*1024
static constexpr size_t OFF_T2    = OFF_T1    + 2097152;
static constexpr size_t OFF_T3    = OFF_T2    + 2097152;
static constexpr size_t OFF_HT    = OFF_T3    + 2097152;     // 2048*512
static constexpr size_t OFF_DH0   = OFF_HT    + 1048576;
static constexpr size_t OFF_DH1   = OFF_DH0   + 1048576;
static constexpr size_t OFF_Z     = OFF_DH1   + 1048576;
static constexpr size_t OFF_G     = OFF_Z     + 1048576;
static constexpr size_t OFF_HF    = OFF_G     + 1048576;
static constexpr size_t OFF_HD    = OFF_HF    + 1048576;
static constexpr size_t OFF_M1    = OFF_HD    + 1048576;     // 2048*256
static constexpr size_t OFF_M2    = OFF_M1    + 524288;
static constexpr size_t OFF_M3    = OFF_M2    + 524288;

extern "C" void kernel_launch(void* const* d_in, const int* in_sizes, int n_in,
                              void* d_out, int out_size, void* d_ws, size_t ws_size,
                              hipStream_t stream) {
    (void)in_sizes; (void)n_in; (void)out_size; (void)ws_size;
    hipStream_t s = stream;
    float* ws = (float*)d_ws;
    float* out = (float*)d_out;

    const int* sf = (const int*)d_in[I_GF];          const int* tf = sf + NE;
    const int* sd = (const int*)d_in[I_GD];          const int* td = sd + NE;
    const float* fmri = (const float*)d_in[I_FMRI];
    const float* dti  = (const float*)d_in[I_DTI];

    float* rs_f_out = ws + OFF_RS;
    float* rs_f_in  = rs_f_out + NN;
    float* rs_d_out = rs_f_in + NN;
    float* rs_d_in  = rs_d_out + NN;
    float* EL = ws + OFF_EL;  float* ER = ws + OFF_ER;  float* MX = ws + OFF_MX;
    float* EMAX = ws + OFF_EMAX; float* DEN = ws + OFF_DEN; float* EE = ws + OFF_EE;
    float* XW = ws + OFF_XW;  float* FM = ws + OFF_FMRIC; float* DT = ws + OFF_DTIC;
    float* P  = ws + OFF_P;   float* S0 = ws + OFF_S0;
    float* T1 = ws + OFF_T1;  float* T2 = ws + OFF_T2;   float* T3 = ws + OFF_T3;
    float* HT = ws + OFF_HT;  float* DH0 = ws + OFF_DH0; float* DH1 = ws + OFF_DH1;
    float* Z  = ws + OFF_Z;   float* G  = ws + OFF_G;
    float* HF = ws + OFF_HF;  float* HD = ws + OFF_HD;
    float* M1 = ws + OFF_M1;  float* M2 = ws + OFF_M2;   float* M3 = ws + OFF_M3;

    const int nbNN = NN * NN / 256;

    // ---- degrees -> rsqrt(max(deg,1)) ----
    fill_f32_kernel<<<4 * NN / 256, 256, 0, s>>>(rs_f_out, 0.f, 4 * NN);
    degree_kernel<<<NE / 256, 256, 0, s>>>(sf, tf, rs_f_out, rs_f_in, NE);
    degree_kernel<<<NE / 256, 256, 0, s>>>(sd, td, rs_d_out, rs_d_in, NE);
    rsqrt_clamp_kernel<<<4 * NN / 256, 256, 0, s>>>(rs_f_out, 4 * NN);

    // ---- graphconv fmri ----
    gemm(s, fmri, (const float*)d_in[I_GCNF_W], XW, NN, NN, NN, 0, nullptr, nullptr, nullptr);
    fill_f32_kernel<<<nbNN, 256, 0, s>>>(FM, 0.f, NN * NN);
    gcn_scatter_kernel<<<NE, 256, 0, s>>>(sf, tf, XW, rs_f_out, FM);
    gcn_finalize_kernel<<<nbNN, 256, 0, s>>>(FM, rs_f_in, (const float*)d_in[I_GCNF_B]);

    // ---- graphconv dti ----
    gemm(s, dti, (const float*)d_in[I_GCND_W], XW, NN, NN, NN, 0, nullptr, nullptr, nullptr);
    fill_f32_kernel<<<nbNN, 256, 0, s>>>(DT, 0.f, NN * NN);
    gcn_scatter_kernel<<<NE, 256, 0, s>>>(sd, td, XW, rs_d_out, DT);
    gcn_finalize_kernel<<<nbNN, 256, 0, s>>>(DT, rs_d_in, (const float*)d_in[I_GCND_B]);

    // ---- S0 = clip(0.5*(FM@DT^T + transpose)); S_G = 0.5*(FM+DT) ----
    gemm(s, FM, DT, P, NN, NN, NN, 1, nullptr, nullptr, nullptr);
    symm_clip_kernel<<<nbNN, 256, 0, s>>>(P, S0);
    avg_kernel<<<nbNN, 256, 0, s>>>(FM, DT, out + 3145728, NN * NN);   // S_G slice

    // ---- mlp_in_t ----
    run_mlp(s, mlp_from(d_in, I_MT), S0, 2048, 1024, 512, T1, T2, T3, HT);

    // ---- down GAT layers (dense) ----
    run_gat_dense(s, HT,  d_in, I_D0, Z, EL, ER, MX, DH0);
    run_gat_dense(s, DH0, d_in, I_D1, Z, EL, ER, MX, DH1);

    // ---- mlp_middle ----
    run_mlp(s, mlp_from(d_in, I_MM), DH1, 512, 512, 512, T1, T2, T3, HT);

    // ---- up 0: h_t += down_hidden[1]; dense GAT up[0] ----
    add_kernel<<<NN * HIDD / 256, 256, 0, s>>>(HT, DH1, NN * HIDD);
    run_gat_dense(s, HT, d_in, I_U0, Z, EL, ER, MX, G);

    // ---- up 1: h_t += down_hidden[0]; sparse GAT up[1] on both graphs ----
    add_kernel<<<NN * HIDD / 256, 256, 0, s>>>(G, DH0, NN * HIDD);
    gemm(s, G, (const float*)d_in[I_U1 + 0], Z, NN, HIDD, HIDD, 0, nullptr, nullptr, nullptr);
    el_er_kernel<<<NN * NH / 8, 256, 0, s>>>(Z, (const float*)d_in[I_U1 + 1],
                                             (const float*)d_in[I_U1 + 2], EL, ER);
    const float* bU1 = (const float*)d_in[I_U1 + 3];
    run_gat_sparse(s, sf, tf, Z, EL, ER, bU1, EMAX, DEN, EE, HF);
    run_gat_sparse(s, sd, td, Z, EL, ER, bU1, EMAX, DEN, EE, HD);

    // ---- output heads ----
    run_mlp(s, mlp_from(d_in, I_MOF), HF, 512, 256, 256, M1, M2, M3, out);            // final_f
    run_mlp(s, mlp_from(d_in, I_MOD), HD, 512, 256, 256, M1, M2, M3, out + 524288);   // final_d
    concat_kernel<<<NN * HIDD / 256, 256, 0, s>>>(HF, HD, out + 1048576);             // out_hidden
}